// SingleStageDetector_72962904424834
// MI455X (gfx1250) — compile-verified
//
#include <hip/hip_runtime.h>
#include <hip/hip_bf16.h>

typedef float v2f __attribute__((ext_vector_type(2)));
typedef float v8f __attribute__((ext_vector_type(8)));

#define Bn   32
#define An   9
#define Hn   64
#define Wn   64
#define Nn   40
#define Mn   256
#define INC  256
#define HIDC 256
#define OUTC 65            // 5*A + C = 45 + 20
#define HWn  4096          // H*W
#define AHW  36864         // A*H*W
#define IOU_ELEMS ((size_t)Bn * (size_t)AHW * (size_t)Nn)

#define XSTR 260           // LDS row stride (floats): even (float2 align) and 4 mod 64 (bank spread)
#define OSTR 68

// ---------------------------------------------------------------------------
// Kernel 1: proposal decode + IoU matrix (bandwidth-bound, ~189 MB out)
// ---------------------------------------------------------------------------
__global__ __launch_bounds__(256) void iou_kernel(
    const float* __restrict__ anc, const float* __restrict__ grid,
    const float* __restrict__ offsets, const float* __restrict__ bboxes,
    float* __restrict__ out)
{
    __shared__ float gb[Nn * 5];   // g0,g1,g2,g3,area_b per box

    const int b     = blockIdx.x / 144;                 // 144 blocks per batch
    const int local = (blockIdx.x % 144) * 256 + threadIdx.x;   // 0..36863

    if (threadIdx.x < Nn) {
        const int n = threadIdx.x;
        const float g0 = bboxes[(b * Nn + n) * 5 + 0];
        const float g1 = bboxes[(b * Nn + n) * 5 + 1];
        const float g2 = bboxes[(b * Nn + n) * 5 + 2];
        const float g3 = bboxes[(b * Nn + n) * 5 + 3];
        gb[n * 5 + 0] = g0; gb[n * 5 + 1] = g1;
        gb[n * 5 + 2] = g2; gb[n * 5 + 3] = g3;
        gb[n * 5 + 4] = (g2 - g0) * (g3 - g1);
    }
    __syncthreads();

    const int a  = local / HWn;
    const int hw = local - a * HWn;

    const float xc = grid[((size_t)b * HWn + hw) * 2 + 0];
    const float yc = grid[((size_t)b * HWn + hw) * 2 + 1];
    const float aw = anc[a * 2 + 0];    // full anchor width  (= wa in ref)
    const float ah = anc[a * 2 + 1];    // full anchor height (= ha in ref)

    const float4 off = *(const float4*)(offsets + (((size_t)b * An + a) * HWn + hw) * 4);
    const float tx = fminf(fmaxf(off.x, -0.5f), 0.5f);
    const float ty = fminf(fmaxf(off.y, -0.5f), 0.5f);
    const float pw = aw * expf(off.z);
    const float ph = ah * expf(off.w);
    const float pcx = xc + tx, pcy = yc + ty;
    const float p0 = pcx - pw * 0.5f, p1 = pcy - ph * 0.5f;
    const float p2 = pcx + pw * 0.5f, p3 = pcy + ph * 0.5f;
    const float area_p = pw * ph;

    const size_t obase = ((size_t)b * AHW + (size_t)local) * Nn;
    for (int n0 = 0; n0 < Nn; n0 += 4) {
        float4 r;
        float* rp = &r.x;
        #pragma unroll
        for (int j = 0; j < 4; ++j) {
            const float* g = &gb[(n0 + j) * 5];
            const float xi = fminf(p2, g[2]) - fmaxf(p0, g[0]);
            const float yi = fminf(p3, g[3]) - fmaxf(p1, g[1]);
            const float inter = fmaxf(xi, 0.0f) * fmaxf(yi, 0.0f);
            rp[j] = inter / (area_p + g[4] - inter);
        }
        *(float4*)(out + obase + n0) = r;
    }
}

// ---------------------------------------------------------------------------
// Kernel 2: gathered-pixel MLP via f32 WMMA + per-entry loss terms
//   32 blocks x 16 entries; block = 256 threads = 8 wave32
//   Layer1: H1(256x16) = leaky(W1(256x256) @ Xg(256x16) + b1)   (WMMA f32 16x16x4)
//   Layer2: O(65x16)   = W2(65x256) @ H1 + b2                   (WMMA f32 16x16x4)
// ---------------------------------------------------------------------------
__global__ __launch_bounds__(256) void mlp_loss_kernel(
    const float* __restrict__ features, const float* __restrict__ w1,
    const float* __restrict__ b1, const float* __restrict__ w2,
    const float* __restrict__ b2, const float* __restrict__ gt,
    const int* __restrict__ pos_idx, const int* __restrict__ neg_idx,
    float* __restrict__ ws)
{
    __shared__ float Xls[16 * XSTR];    // Xls[p][c]  : features of entry p
    __shared__ float H1ls[16 * XSTR];   // H1ls[p][m] : hidden activations
    __shared__ float Outls[16 * OSTR];  // Outls[p][o]: output channels 0..64
    __shared__ int   sA[16];
    __shared__ int   sBase[16];
    __shared__ float sCl[16], sRl[16];

    const int tid = threadIdx.x;
    const int blk = blockIdx.x;

    if (tid < 16) {
        const int e   = blk * 16 + tid;
        const int idx = (e < Mn) ? pos_idx[e] : neg_idx[e - Mn];
        const int b   = idx / AHW;
        const int rem = idx - b * AHW;
        const int a   = rem / HWn;
        const int hw  = rem - a * HWn;
        sA[tid]    = a;
        sBase[tid] = b * (INC * HWn) + hw;
    }
    __syncthreads();

    // gather: thread t -> entry p = t>>4, channels (t&15) + 16*j
    {
        const int p  = tid >> 4;
        const int c0 = tid & 15;
        const int base = sBase[p];
        #pragma unroll
        for (int j = 0; j < 16; ++j) {
            const int c = c0 + j * 16;
            Xls[p * XSTR + c] = features[(size_t)base + (size_t)c * HWn];
        }
    }
    __syncthreads();

    const int wave = tid >> 5;
    const int lane = tid & 31;
    const int half = lane >> 4;     // K-half selector per ISA layout
    const int l    = lane & 15;     // M (A-frag) / N (B-frag, C/D) lane index

    // ---- Layer 1: each wave covers M-tiles {wave, wave+8} ----
    for (int t = 0; t < 2; ++t) {
        const int mbase = (wave + t * 8) * 16;
        v8f c = {};
        const float* arow = w1 + (size_t)(mbase + l) * INC + 2 * half;
        const float* brow = Xls + l * XSTR + 2 * half;
        #pragma unroll 4
        for (int k0 = 0; k0 < INC; k0 += 4) {
            v2f av = *(const v2f*)(arow + k0);
            v2f bv = *(const v2f*)(brow + k0);
            c = __builtin_amdgcn_wmma_f32_16x16x4_f32(
                    false, av, false, bv, (short)0, c, false, false);
        }
        #pragma unroll
        for (int j = 0; j < 8; ++j) {
            const int m = mbase + j + 8 * half;
            float h = c[j] + b1[m];
            h = (h > 0.0f) ? h : 0.01f * h;     // leaky_relu, slope 0.01
            H1ls[l * XSTR + m] = h;
        }
    }
    __syncthreads();

    // ---- Layer 2: 5 M-tiles cover 65 rows; waves 5..7 duplicate tiles 0..2
    //      (identical writes) so EXEC is all-ones for every WMMA. ----
    {
        const int mtile = (wave < 5) ? wave : (wave - 5);
        const int mbase = mtile * 16;
        const int row   = mbase + l;
        const bool rowok = (row < OUTC);
        v8f c = {};
        const float* arow = w2 + (size_t)row * HIDC + 2 * half;
        const float* brow = H1ls + l * XSTR + 2 * half;
        #pragma unroll 4
        for (int k0 = 0; k0 < HIDC; k0 += 4) {
            v2f av;
            if (rowok) { av = *(const v2f*)(arow + k0); }
            else       { av.x = 0.0f; av.y = 0.0f; }
            v2f bv = *(const v2f*)(brow + k0);
            c = __builtin_amdgcn_wmma_f32_16x16x4_f32(
                    false, av, false, bv, (short)0, c, false, false);
        }
        #pragma unroll
        for (int j = 0; j < 8; ++j) {
            const int m = mbase + j + 8 * half;
            if (m < OUTC) Outls[l * OSTR + m] = c[j] + b2[m];
        }
    }
    __syncthreads();

    // ---- per-entry loss terms (deterministic fixed-order block reduce) ----
    if (tid < 16) {
        const int p = tid;
        const int e = blk * 16 + p;
        const int a = sA[p];
        const float conf = Outls[p * OSTR + a];
        const float tgt  = (e < Mn) ? 1.0f : 0.0f;
        const float d    = conf - tgt;
        float rl = 0.0f;
        if (e < Mn) {
            #pragma unroll
            for (int j = 0; j < 4; ++j) {
                const float dv = Outls[p * OSTR + An + 4 * a + j] - gt[e * 4 + j];
                rl += dv * dv;
            }
        }
        sCl[p] = d * d;
        sRl[p] = rl;
    }
    __syncthreads();
    if (tid == 0) {
        float cs = 0.0f, rs = 0.0f;
        for (int p = 0; p < 16; ++p) { cs += sCl[p]; rs += sRl[p]; }
        ws[2 * blk]     = cs;
        ws[2 * blk + 1] = rs;
    }
}

// ---------------------------------------------------------------------------
// Kernel 3: fixed-order final reduction of 32 block partials
// ---------------------------------------------------------------------------
__global__ void loss_finalize_kernel(const float* __restrict__ ws,
                                     float* __restrict__ out_loss)
{
    if (threadIdx.x == 0 && blockIdx.x == 0) {
        float cs = 0.0f, rs = 0.0f;
        for (int i = 0; i < 32; ++i) { cs += ws[2 * i]; rs += ws[2 * i + 1]; }
        out_loss[0] = cs / (2.0f * (float)Mn);
        out_loss[1] = rs / (float)Mn;
    }
}

extern "C" void kernel_launch(void* const* d_in, const int* in_sizes, int n_in,
                              void* d_out, int out_size, void* d_ws, size_t ws_size,
                              hipStream_t stream) {
    (void)in_sizes; (void)n_in; (void)out_size; (void)ws_size;
    const float* anc      = (const float*)d_in[0];
    const float* grid     = (const float*)d_in[1];
    const float* offsets  = (const float*)d_in[2];
    const float* bboxes   = (const float*)d_in[3];
    const float* features = (const float*)d_in[4];
    const float* w1       = (const float*)d_in[5];
    const float* b1       = (const float*)d_in[6];
    const float* w2       = (const float*)d_in[7];
    const float* b2       = (const float*)d_in[8];
    const float* gt       = (const float*)d_in[9];
    const int*   pos_idx  = (const int*)d_in[10];
    const int*   neg_idx  = (const int*)d_in[11];
    float* out = (float*)d_out;
    float* ws  = (float*)d_ws;

    iou_kernel<<<Bn * (AHW / 256), 256, 0, stream>>>(anc, grid, offsets, bboxes, out);
    mlp_loss_kernel<<<(2 * Mn) / 16, 256, 0, stream>>>(features, w1, b1, w2, b2, gt,
                                                       pos_idx, neg_idx, ws);
    loss_finalize_kernel<<<1, 32, 0, stream>>>(ws, out + IOU_ELEMS);
}